// LSM_24094766531324
// MI455X (gfx1250) — compile-verified
//
#include <hip/hip_runtime.h>
#include <math.h>

typedef float v2f __attribute__((ext_vector_type(2)));
typedef float v8f __attribute__((ext_vector_type(8)));

#define EPS_F 1e-6f
#define GY 32   // j-slices in grid.y

// (log2 e)^2 scaling folded into the quadratic form so the kernel computes
// d2' = L2 * dist^2, giving exp(-dist) == exp2(-sqrt(d2')) with no per-element
// multiply by log2(e).
#define L2_F   2.08136898100584745f   // (log2 e)^2
#define M2L2_F -4.16273796201169490f  // -2 * (log2 e)^2

// ---------------------------------------------------------------------------
// Pass 1: build padded side arrays.
//   Row side : a0,a1 = z + EPS ; suL = L2*|z+EPS|^2 ; eg = exp(gamma)
//   Col side : colpk = {z0, z1, L2*|z|^2, exp(gamma)}  (one b128 load/tile)
//   Padding (i >= N) gets eg = 0, so it contributes exactly zero downstream.
// ---------------------------------------------------------------------------
__global__ __launch_bounds__(256) void lsm_prep_kernel(
    const float* __restrict__ Z, const float* __restrict__ gamma,
    int N, int Npad,
    float4* __restrict__ colpk,
    float* __restrict__ a0, float* __restrict__ a1,
    float* __restrict__ suL, float* __restrict__ eg) {
  int i = blockIdx.x * blockDim.x + threadIdx.x;
  if (i >= Npad) return;
  float z0 = 0.f, z1 = 0.f, e = 0.f;
  if (i < N) {
    z0 = Z[2 * i];
    z1 = Z[2 * i + 1];
    e  = __expf(gamma[i]);
  }
  float u0 = z0 + EPS_F, u1 = z1 + EPS_F;
  a0[i] = u0;
  a1[i] = u1;
  suL[i] = L2_F * (u0 * u0 + u1 * u1);
  eg[i]  = e;
  colpk[i] = make_float4(z0, z1, L2_F * (z0 * z0 + z1 * z1), e);
}

// ---------------------------------------------------------------------------
// Pass 2: dense quadratic form  sum_{i != j} eg_i * eg_j * exp(-||u_i - v_j||)
// One wave per (i-tile, j-slice). Gram tile via V_WMMA_F32_16X16X4_F32.
//
// A (16x4 f32) fragment: lanes 0-15 hold row M=lane, VGPR0=K0, VGPR1=K1;
// lanes 16-31 hold K2/K3 = zero padding (D=2).
// B (4x16 f32) fragment mirrors A with N=lane.
// C/D (16x16 f32): VGPR r, lanes 0-15 -> (M=r,   N=lane),
//                          lanes16-31 -> (M=r+8, N=lane-16).
//
// Diagonal removal: grow==col can only occur when jt==it (wave-uniform), and
// within that tile only for lanes with l16 == 8*half + r (loop-invariant).
// The hot path (jt != it) carries no masking cost at all.
// ---------------------------------------------------------------------------
__global__ __launch_bounds__(32) void lsm_dense_kernel(
    const float* __restrict__ a0, const float* __restrict__ a1,
    const float* __restrict__ suL, const float* __restrict__ eg,
    const float4* __restrict__ colpk, int NT,
    float* __restrict__ partial) {
  const int it   = blockIdx.x;       // i-tile index
  const int lane = threadIdx.x;      // 0..31 (wave32)
  const int half = lane >> 4;        // 0: K0/K1 data, 1: K2/K3 zero-pad
  const int l16  = lane & 15;

  // A fragment for this i-tile (fixed across the j loop)
  const int arow = it * 16 + l16;
  v2f afrag;
  afrag.x = half ? 0.f : a0[arow];
  afrag.y = half ? 0.f : a1[arow];

  // Row-side constants for the 8 C rows this lane owns (M = rbase + r)
  const int rbase = it * 16 + 8 * half;
  float rowS[8], rowE[8];
  bool  dmask[8];                    // true where this lane's C row is the diag
  const int rd = l16 - 8 * half;
#pragma unroll
  for (int r = 0; r < 8; ++r) {
    rowS[r]  = suL[rbase + r];
    rowE[r]  = eg[rbase + r];
    dmask[r] = (rd == r);
  }

  float acc[8];
#pragma unroll
  for (int r = 0; r < 8; ++r) acc[r] = 0.f;

  for (int jt = blockIdx.y; jt < NT; jt += GY) {
    const int col = jt * 16 + l16;
    const float4 cp = colpk[col];    // {z0, z1, L2*|v|^2, eg}
    v2f bfrag;
    bfrag.x = half ? 0.f : cp.x;
    bfrag.y = half ? 0.f : cp.y;

    v8f c = {0.f, 0.f, 0.f, 0.f, 0.f, 0.f, 0.f, 0.f};
    c = __builtin_amdgcn_wmma_f32_16x16x4_f32(
        /*neg_a=*/false, afrag, /*neg_b=*/false, bfrag,
        /*c_mod=*/(short)0, c, /*reuse_a=*/false, /*reuse_b=*/false);

    const float colS = cp.z;
    const float colE = cp.w;
    float w[8];
#pragma unroll
    for (int r = 0; r < 8; ++r) {
      float d2 = fmaf(M2L2_F, c[r], rowS[r] + colS);   // L2 * dist^2
      d2 = fmaxf(d2, 0.f);
      // raw v_sqrt_f32; negate folds into v_exp_f32 src modifier
      const float k = __builtin_amdgcn_exp2f(-__builtin_amdgcn_sqrtf(d2));
      w[r] = rowE[r] * colE * k;     // padded rows/cols: eg==0 -> 0
    }
    if (jt != it) {                  // uniform branch: hot path, no masking
#pragma unroll
      for (int r = 0; r < 8; ++r) acc[r] += w[r];
    } else {                         // diagonal tile: exact K[i,i]=0 removal
#pragma unroll
      for (int r = 0; r < 8; ++r) acc[r] += dmask[r] ? 0.f : w[r];
    }
  }

  float accs = ((acc[0] + acc[1]) + (acc[2] + acc[3])) +
               ((acc[4] + acc[5]) + (acc[6] + acc[7]));

  // Deterministic wave32 tree reduction
#pragma unroll
  for (int off = 16; off > 0; off >>= 1)
    accs += __shfl_down(accs, off, 32);
  if (lane == 0) partial[blockIdx.x * GY + blockIdx.y] = accs;
}

// ---------------------------------------------------------------------------
// Pass 3: edge (link) term: sum over edges of (-||z_i - z_j + eps|| + g_i + g_j)
// ---------------------------------------------------------------------------
__global__ __launch_bounds__(256) void lsm_edge_kernel(
    const float* __restrict__ Z, const float* __restrict__ gamma,
    const int* __restrict__ ei, const int* __restrict__ ej,
    int E, float* __restrict__ partial) {
  __shared__ float s[256];
  const int t = threadIdx.x;
  const int e = blockIdx.x * 256 + t;
  float val = 0.f;
  if (e < E) {
    const int i = ei[e], j = ej[e];
    const float d0 = Z[2 * i]     - Z[2 * j]     + EPS_F;
    const float d1 = Z[2 * i + 1] - Z[2 * j + 1] + EPS_F;
    const float dist = __builtin_amdgcn_sqrtf(d0 * d0 + d1 * d1);
    val = -dist + gamma[i] + gamma[j];
  }
  s[t] = val;
  __syncthreads();
  for (int off = 128; off > 0; off >>= 1) {
    if (t < off) s[t] += s[t + off];
    __syncthreads();
  }
  if (t == 0) partial[blockIdx.x] = s[0];
}

// ---------------------------------------------------------------------------
// Pass 4: final deterministic reduction (fixed order, double accumulation)
//   out = link_sum - 0.5 * quad_sum
// ---------------------------------------------------------------------------
__global__ __launch_bounds__(256) void lsm_final_kernel(
    const float* __restrict__ denseP, int nD,
    const float* __restrict__ edgeP, int nE,
    float* __restrict__ out) {
  __shared__ double s[256];
  const int t = threadIdx.x;
  double v = 0.0;
  for (int k = t; k < nE; k += 256) v += (double)edgeP[k];
  for (int k = t; k < nD; k += 256) v -= 0.5 * (double)denseP[k];
  s[t] = v;
  __syncthreads();
  for (int off = 128; off > 0; off >>= 1) {
    if (t < off) s[t] += s[t + off];
    __syncthreads();
  }
  if (t == 0) out[0] = (float)s[0];
}

// ---------------------------------------------------------------------------
extern "C" void kernel_launch(void* const* d_in, const int* in_sizes, int n_in,
                              void* d_out, int out_size, void* d_ws, size_t ws_size,
                              hipStream_t stream) {
  const float* Z     = (const float*)d_in[0];
  const float* gamma = (const float*)d_in[1];
  const int*   ei    = (const int*)d_in[2];
  const int*   ej    = (const int*)d_in[3];

  const int N    = in_sizes[1];          // gamma length
  const int E    = in_sizes[2];
  const int NT   = (N + 15) / 16;        // 16x16 tiles per side
  const int Npad = NT * 16;

  // Workspace layout (float4 array first to keep 16B alignment)
  float4* colpk = (float4*)d_ws;
  float* fb = (float*)(colpk + Npad);
  float* a0  = fb;
  float* a1  = a0 + Npad;
  float* suL = a1 + Npad;
  float* eg  = suL + Npad;
  float* denseP = eg + Npad;
  const int nD = NT * GY;
  float* edgeP = denseP + nD;
  const int nE = (E + 255) / 256;

  lsm_prep_kernel<<<(Npad + 255) / 256, 256, 0, stream>>>(
      Z, gamma, N, Npad, colpk, a0, a1, suL, eg);

  lsm_dense_kernel<<<dim3(NT, GY), 32, 0, stream>>>(
      a0, a1, suL, eg, colpk, NT, denseP);

  lsm_edge_kernel<<<nE, 256, 0, stream>>>(Z, gamma, ei, ej, E, edgeP);

  lsm_final_kernel<<<1, 256, 0, stream>>>(denseP, nD, edgeP, nE, (float*)d_out);
}